// GTConvAE_45509473469014
// MI455X (gfx1250) — compile-verified
//
#include <hip/hip_runtime.h>

typedef __attribute__((ext_vector_type(2))) float v2f;
typedef __attribute__((ext_vector_type(8))) float v8f;

// ---------------------------------------------------------------------------
// G = Sg (192x192) @ Z (192 x TC), all row-major f32, TC compile-time.
// One wave32 per 16x32 output tile: two v8f accumulators share each A
// fragment, K=192 accumulated as 48 x 2 V_WMMA_F32_16X16X4_F32.
// A-frag (16x4, MxK): lanes 0-15 hold K={0,1} in VGPR{0,1}, lanes 16-31 hold
// K={2,3}.  B-frag (4x16, KxN): mirrored.  C/D (16x16 f32): VGPR v holds
// row v (lanes 0-15) / row v+8 (lanes 16-31).
// TC is a template constant so every global access folds into an immediate
// instruction offset (24-bit IOFFSET covers 192*512*4B) instead of u64 VALU
// address chains.
// ---------------------------------------------------------------------------
template <int TC>
__global__ __launch_bounds__(32) void sg_apply_wmma(
    const float* __restrict__ Sg, const float* __restrict__ Z,
    float* __restrict__ G) {
  const int nt    = blockIdx.x;        // 32-wide column tile
  const int mt    = blockIdx.y;        // 16-tall row tile
  const int l     = threadIdx.x;       // 0..31
  const int half  = l >> 4;            // 0 -> K pair {0,1}; 1 -> K pair {2,3}
  const int lm    = l & 15;
  const int arow  = mt * 16 + lm;      // A row owned by this lane
  const int bcol0 = nt * 32 + lm;      // first B/C column owned by this lane
  const int bcol1 = bcol0 + 16;        // second B/C column

  v8f acc0 = {0.f, 0.f, 0.f, 0.f, 0.f, 0.f, 0.f, 0.f};
  v8f acc1 = {0.f, 0.f, 0.f, 0.f, 0.f, 0.f, 0.f, 0.f};

  const float* __restrict__ ap  = Sg + arow * 192 + half * 2;
  const float* __restrict__ bp0 = Z + half * 2 * TC + bcol0;
  const float* __restrict__ bp1 = Z + half * 2 * TC + bcol1;

#pragma unroll
  for (int k0 = 0; k0 < 192; k0 += 4) {
    v2f a, b0, b1;
    a.x  = ap[k0];
    a.y  = ap[k0 + 1];
    b0.x = bp0[k0 * TC];
    b0.y = bp0[k0 * TC + TC];
    b1.x = bp1[k0 * TC];
    b1.y = bp1[k0 * TC + TC];
    acc0 = __builtin_amdgcn_wmma_f32_16x16x4_f32(
        false, a, false, b0, (short)0, acc0, false, false);
    acc1 = __builtin_amdgcn_wmma_f32_16x16x4_f32(
        false, a, false, b1, (short)0, acc1, false, false);
  }

  const int rbase = mt * 16 + half * 8;
  float* __restrict__ gp = G + rbase * TC;
#pragma unroll
  for (int v = 0; v < 8; ++v) {
    gp[v * TC + bcol0] = acc0[v];
    gp[v * TC + bcol1] = acc1[v];
  }
}

// ---------------------------------------------------------------------------
// y = s00*x + s01*g + s10*shift(x) + s11*shift(g), cyclic shift over time.
// Layout [node][t][c]; linear idx == (node*t + tau)*c + ch.
// ---------------------------------------------------------------------------
__global__ void combine_shift(const float* __restrict__ x,
                              const float* __restrict__ g,
                              const float* __restrict__ s,
                              float* __restrict__ y, int t, int c) {
  const int idx = blockIdx.x * blockDim.x + threadIdx.x;
  const int total = 192 * t * c;
  if (idx >= total) return;
  const int ch   = idx % c;
  const int tmp  = idx / c;
  const int tau  = tmp % t;
  const int node = tmp / t;
  const int taum = (tau + t - 1) % t;
  const int i1 = (node * t + taum) * c + ch;
  const float s00 = s[0], s01 = s[1], s10 = s[2], s11 = s[3];
  y[idx] = s00 * x[idx] + s01 * g[idx] + s10 * x[i1] + s11 * g[i1];
}

// ---------------------------------------------------------------------------
// Y[node,tau,o] = sum_k sum_i h[o,i,k] * xk[node,tau,i]   (h: [cout][cin][3])
// ---------------------------------------------------------------------------
__global__ void mix_channels(const float* __restrict__ x0,
                             const float* __restrict__ x1,
                             const float* __restrict__ x2,
                             const float* __restrict__ h,
                             float* __restrict__ y,
                             int t, int cin, int cout) {
  const int idx = blockIdx.x * blockDim.x + threadIdx.x;
  const int total = 192 * t * cout;
  if (idx >= total) return;
  const int o    = idx % cout;
  const int tmp  = idx / cout;
  const int tau  = tmp % t;
  const int node = tmp / t;
  const int base = (node * t + tau) * cin;
  float acc = 0.f;
  for (int i = 0; i < cin; ++i) {
    const float* hp = h + (o * cin + i) * 3;
    acc = fmaf(hp[0], x0[base + i], acc);
    acc = fmaf(hp[1], x1[base + i], acc);
    acc = fmaf(hp[2], x2[base + i], acc);
  }
  y[idx] = acc;
}

// ---------------------------------------------------------------------------
// a[node,tau',c] = relu(max(y[node,2tau',c], y[node,2tau'+1,c]))
// ---------------------------------------------------------------------------
__global__ void down_relu(const float* __restrict__ y, float* __restrict__ a,
                          int t_out, int c) {
  const int idx = blockIdx.x * blockDim.x + threadIdx.x;
  const int total = 192 * t_out * c;
  if (idx >= total) return;
  const int ch   = idx % c;
  const int tmp  = idx / c;
  const int tau  = tmp % t_out;
  const int node = tmp / t_out;
  const int b0 = (node * 2 * t_out + 2 * tau) * c + ch;
  const float m = fmaxf(y[b0], y[b0 + c]);
  a[idx] = fmaxf(m, 0.f);
}

// ---------------------------------------------------------------------------
// a[node,tau,c] = (tau even) ? relu(y[node,tau/2,c]) : 0   (zero-stuff x2)
// ---------------------------------------------------------------------------
__global__ void up_relu(const float* __restrict__ y, float* __restrict__ a,
                        int t_in, int c) {
  const int idx = blockIdx.x * blockDim.x + threadIdx.x;
  const int t_out = 2 * t_in;
  const int total = 192 * t_out * c;
  if (idx >= total) return;
  const int ch   = idx % c;
  const int tmp  = idx / c;
  const int tau  = tmp % t_out;
  const int node = tmp / t_out;
  float v = 0.f;
  if ((tau & 1) == 0) v = fmaxf(y[(node * t_in + (tau >> 1)) * c + ch], 0.f);
  a[idx] = v;
}

// ---------------------------------------------------------------------------
// Dispatch the templated WMMA matmul for the three occurring tc widths.
// ---------------------------------------------------------------------------
static void launch_sg(const float* Sg, const float* Z, float* G, int tc,
                      hipStream_t stream) {
  const dim3 grid(tc / 32, 12);
  switch (tc) {
    case 32:
      sg_apply_wmma<32><<<grid, 32, 0, stream>>>(Sg, Z, G);
      break;
    case 256:
      sg_apply_wmma<256><<<grid, 32, 0, stream>>>(Sg, Z, G);
      break;
    default:
      sg_apply_wmma<512><<<grid, 32, 0, stream>>>(Sg, Z, G);
      break;
  }
}

// ---------------------------------------------------------------------------
// One GTConv layer: x1 = S x0, x2 = S x1 (each S-apply = Sg-WMMA + shift
// combine), then channel/tap mixing.  Y may alias d_out on the last layer.
// ---------------------------------------------------------------------------
static void conv_layer(const float* Sg, const float* s, const float* x0,
                       const float* h, float* X1, float* X2, float* G,
                       float* Y, int t, int cin, int cout, hipStream_t stream) {
  const int tc   = t * cin;
  const int totc = 192 * t * cin;
  const int toty = 192 * t * cout;
  launch_sg(Sg, x0, G, tc, stream);
  combine_shift<<<(totc + 255) / 256, 256, 0, stream>>>(x0, G, s, X1, t, cin);
  launch_sg(Sg, X1, G, tc, stream);
  combine_shift<<<(totc + 255) / 256, 256, 0, stream>>>(X1, G, s, X2, t, cin);
  mix_channels<<<(toty + 255) / 256, 256, 0, stream>>>(x0, X1, X2, h, Y, t,
                                                       cin, cout);
}

extern "C" void kernel_launch(void* const* d_in, const int* in_sizes, int n_in,
                              void* d_out, int out_size, void* d_ws,
                              size_t ws_size, hipStream_t stream) {
  (void)in_sizes; (void)n_in; (void)out_size; (void)ws_size;
  const float* X    = (const float*)d_in[0];  // (192, 32) == [node][tau]
  const float* Sg   = (const float*)d_in[1];  // (192, 192)
  const float* s    = (const float*)d_in[2];  // (2, 2) row-major
  const float* h_e1 = (const float*)d_in[3];  // (16, 1, 3)
  const float* h_e2 = (const float*)d_in[4];  // (32, 16, 3)
  const float* h_d1 = (const float*)d_in[5];  // (16, 32, 3)
  const float* h_d2 = (const float*)d_in[6];  // (1, 16, 3)
  float* out = (float*)d_out;                 // (192, 32)
  float* ws  = (float*)d_ws;

  const int BUF = 192 * 32 * 16;  // 98304 floats: max activation size
  float* X1 = ws + 0 * BUF;
  float* X2 = ws + 1 * BUF;
  float* G  = ws + 2 * BUF;
  float* Y  = ws + 3 * BUF;
  float* A0 = ws + 4 * BUF;
  float* A1 = ws + 5 * BUF;

  // ---- encoder layer 1: t=32, 1 -> 16 channels (Sp0), maxpool2 + relu ----
  conv_layer(Sg, s, X, h_e1, X1, X2, G, Y, 32, 1, 16, stream);
  down_relu<<<(192 * 16 * 16 + 255) / 256, 256, 0, stream>>>(Y, A0, 16, 16);

  // ---- encoder layer 2: t=16, 16 -> 32 channels (Sp1), maxpool2 + relu ----
  conv_layer(Sg, s, A0, h_e2, X1, X2, G, Y, 16, 16, 32, stream);
  down_relu<<<(192 * 8 * 32 + 255) / 256, 256, 0, stream>>>(Y, A1, 8, 32);

  // ---- decoder layer 1: zero-stuff 8->16, relu, conv t=16, 32 -> 16 ----
  up_relu<<<(192 * 16 * 32 + 255) / 256, 256, 0, stream>>>(A1, A0, 8, 32);
  conv_layer(Sg, s, A0, h_d1, X1, X2, G, Y, 16, 32, 16, stream);

  // ---- decoder layer 2: zero-stuff 16->32, relu, conv t=32, 16 -> 1 ----
  up_relu<<<(192 * 32 * 16 + 255) / 256, 256, 0, stream>>>(Y, A1, 16, 16);
  conv_layer(Sg, s, A1, h_d2, X1, X2, G, out, 32, 16, 1, stream);
}